// SparseMoEDispatcher_73100343378254
// MI455X (gfx1250) — compile-verified
//
#include <hip/hip_runtime.h>
#include <math.h>

typedef __attribute__((ext_vector_type(2))) float v2f;
typedef __attribute__((ext_vector_type(8))) float v8f;
typedef int vi2 __attribute__((vector_size(8)));    // matches builtin's V2i
typedef int vi4 __attribute__((vector_size(16)));   // matches builtin's V4i

// ---- problem dims (match reference) ----
constexpr int E_   = 8;      // experts
constexpr int TOPK = 2;
constexpr int Dm   = 1024;   // d_model
constexpr int Ff   = 2048;   // d_ff
constexpr int T_   = 2048;   // tokens (B*S)

// ---- tiling ----
constexpr int MTILE = 32;    // rows (token-slots) per block
constexpr int NTILE = 512;   // cols per block (8 waves x 64)
constexpr int KC    = 16;    // k-chunk staged in LDS

constexpr int MAXSLOTS   = T_ * TOPK + E_ * MTILE;   // 4096 + 256 = 4352
constexpr int MAX_MTILES = MAXSLOTS / MTILE;         // 136

// ---- workspace layout (bytes) ----
constexpr size_t WS_BASE = 0;                                   // int[16]
constexpr size_t WS_S2T  = 256;                                 // int[MAXSLOTS]
constexpr size_t WS_T2S  = ((WS_S2T + (size_t)MAXSLOTS*4 + 255)/256)*256;   // int[T_*TOPK]
constexpr size_t WS_TW   = ((WS_T2S + (size_t)T_*TOPK*4 + 255)/256)*256;    // float[T_*TOPK]
constexpr size_t WS_H    = ((WS_TW + (size_t)T_*TOPK*4 + 65535)/65536)*65536; // float[MAXSLOTS*Ff]
constexpr size_t WS_Y    = WS_H + (size_t)MAXSLOTS * Ff * 4;                  // float[MAXSLOTS*Dm]

// ---- CDNA5 async global->LDS copy (ASYNCcnt-tracked), with sync fallback ----
#if __has_builtin(__builtin_amdgcn_global_load_async_to_lds_b128) && \
    __has_builtin(__builtin_amdgcn_global_load_async_to_lds_b64)
#define USE_ASYNC 1
#else
#define USE_ASYNC 0
#endif

// exact-typed address-space casts for the async builtins
#define GPTR2(p) ((__attribute__((address_space(1))) vi2*)(unsigned long long)(p))
#define GPTR4(p) ((__attribute__((address_space(1))) vi4*)(unsigned long long)(p))
#define LPTR2(p) ((__attribute__((address_space(3))) vi2*)(unsigned int)(unsigned long long)(p))
#define LPTR4(p) ((__attribute__((address_space(3))) vi4*)(unsigned int)(unsigned long long)(p))

// =====================================================================
// Router: softmax -> top-2 -> renorm weights -> compacted expert lists
// =====================================================================
__global__ __launch_bounds__(1024)
void moe_router(const float* __restrict__ logits,
                int* __restrict__ base, int* __restrict__ s2t,
                int* __restrict__ t2s, float* __restrict__ tw)
{
    __shared__ int cnt[E_];
    __shared__ int cur[E_];
    __shared__ int sbase[E_ + 1];
    const int tid = threadIdx.x;
    if (tid < E_) { cnt[tid] = 0; cur[tid] = 0; }
    for (int s = tid; s < MAXSLOTS; s += 1024) s2t[s] = -1;
    __syncthreads();

    int   eidx[2][TOPK];
    float ew[2][TOPK];
    for (int r = 0; r < T_ / 1024; ++r) {
        const int t = tid + r * 1024;
        float p[E_];
        float mx = -1e30f;
        #pragma unroll
        for (int e = 0; e < E_; ++e) { p[e] = logits[t * E_ + e]; mx = fmaxf(mx, p[e]); }
        #pragma unroll
        for (int e = 0; e < E_; ++e) p[e] = __expf(p[e] - mx);   // softmax denom cancels in renorm
        int i0 = 0;
        #pragma unroll
        for (int e = 1; e < E_; ++e) if (p[e] > p[i0]) i0 = e;
        int i1 = (i0 == 0) ? 1 : 0;
        #pragma unroll
        for (int e = 0; e < E_; ++e) if (e != i0 && p[e] > p[i1]) i1 = e;
        const float inv = 1.f / (p[i0] + p[i1]);
        eidx[r][0] = i0; ew[r][0] = p[i0] * inv;
        eidx[r][1] = i1; ew[r][1] = p[i1] * inv;
        atomicAdd(&cnt[i0], 1);
        atomicAdd(&cnt[i1], 1);
    }
    __syncthreads();
    if (tid == 0) {
        int b = 0;
        for (int e = 0; e < E_; ++e) {
            sbase[e] = b;
            b += ((cnt[e] + MTILE - 1) / MTILE) * MTILE;   // 32-align each expert range
        }
        sbase[E_] = b;
        for (int e = 0; e <= E_; ++e) base[e] = sbase[e];
    }
    __syncthreads();
    for (int r = 0; r < T_ / 1024; ++r) {
        const int t = tid + r * 1024;
        #pragma unroll
        for (int k = 0; k < TOPK; ++k) {
            const int e   = eidx[r][k];
            const int pos = atomicAdd(&cur[e], 1);
            const int slot = sbase[e] + pos;
            s2t[slot] = t;
            t2s[t * TOPK + k] = slot;
            tw[t * TOPK + k]  = ew[r][k];
        }
    }
}

// =====================================================================
// Tiled fp32 WMMA GEMM over compacted slots.
//   GATHER=1: A rows gathered from hidden via slot2tok (GEMM1, +silu)
//   GATHER=0: A rows read directly from H (GEMM2)
// Block: 256 thr = 8 waves; tile 32(M) x 512(N); K in 16-chunks.
// Async path: double-buffered GLOBAL_LOAD_ASYNC_TO_LDS + s_wait_asynccnt.
// =====================================================================
template <int KDIM, int NW, bool GATHER, bool SILU>
__global__ __launch_bounds__(256)
void moe_gemm(const float* __restrict__ Amat,   // [*, KDIM]
              const float* __restrict__ Wmat,   // [E, KDIM, NW]
              float* __restrict__ Out,          // [MAXSLOTS, NW]
              const int* __restrict__ base,
              const int* __restrict__ s2t)
{
#if USE_ASYNC
    __shared__ float Alds[2][MTILE][KC + 4];    // [buf][m][k], pad->20
    __shared__ float Blds[2][KC][NTILE + 16];   // [buf][k][n], pad->528
#else
    __shared__ float Alds[1][MTILE][KC + 4];
    __shared__ float Blds[1][KC][NTILE + 16];
#endif

    const int slotBase = blockIdx.y * MTILE;
    if (slotBase >= base[E_]) return;           // uniform early-exit past used slots
    int e = 0;
    #pragma unroll
    for (int i = 1; i < E_; ++i) if (slotBase >= base[i]) e = i;
    const float* Bbase = Wmat + (size_t)e * KDIM * NW + blockIdx.x * NTILE;
    const int nblk = blockIdx.x * NTILE;

    const int wave = threadIdx.x >> 5;
    const int lane = threadIdx.x & 31;
    const int half = lane >> 4;                 // hi half-wave handles K+2 / M+8
    const int ln   = lane & 15;

    const v8f vzero = {0.f, 0.f, 0.f, 0.f, 0.f, 0.f, 0.f, 0.f};
    v8f acc[2][4];
    #pragma unroll
    for (int i = 0; i < 2; ++i)
        #pragma unroll
        for (int j = 0; j < 4; ++j) acc[i][j] = vzero;

    // A staging: thread -> (row m, k-pair); chunk-invariant row index
    const int am = threadIdx.x >> 3;            // 0..31
    const int ak = (threadIdx.x & 7) * 2;       // 0,2,..,14
    int arow;
    if (GATHER) arow = s2t[slotBase + am];      // may be -1 (padding row)
    else        arow = slotBase + am;
    const float* Arow = Amat + (size_t)(arow < 0 ? 0 : arow) * KDIM + ak;

#if USE_ASYNC
    // ---- stage one K-chunk into LDS buffer `buf` (9 async ops per wave) ----
    auto stage = [&](int k0, int buf) {
        if (GATHER && arow < 0) {
            *(v2f*)&Alds[buf][am][ak] = (v2f){0.f, 0.f};          // zero padding row
        } else {
            __builtin_amdgcn_global_load_async_to_lds_b64(
                GPTR2(Arow + k0), LPTR2(&Alds[buf][am][ak]), 0, 0);
        }
        const float* Bp = Bbase + (size_t)k0 * NW;
        #pragma unroll
        for (int j = 0; j < 8; ++j) {
            const int p   = threadIdx.x + 256 * j;
            const int row = p >> 7;                 // 128 float4 per row
            const int c4  = (p & 127) << 2;
            __builtin_amdgcn_global_load_async_to_lds_b128(
                GPTR4(Bp + (size_t)row * NW + c4), LPTR4(&Blds[buf][row][c4]), 0, 0);
        }
    };

    stage(0, 0);
    for (int it = 0; it < KDIM / KC; ++it) {
        const int buf = it & 1;
        // current chunk (this wave's 9 async ops) has landed in LDS
        asm volatile("s_wait_asynccnt 0x0" ::: "memory");
        __syncthreads();     // all waves' copies landed; all done reading buf^1
        if (it + 1 < KDIM / KC) stage((it + 1) * KC, buf ^ 1);  // prefetch next
#else
    for (int it = 0; it < KDIM / KC; ++it) {
        const int buf = 0;
        const int k0s = it * KC;
        __syncthreads();
        {   // A tile (32 x 16)
            v2f av = {0.f, 0.f};
            if (!GATHER || arow >= 0) av = *(const v2f*)(Arow + k0s);
            *(v2f*)&Alds[buf][am][ak] = av;
        }
        {   // B tile (16 x 512), coalesced float4
            const float* Bp = Bbase + (size_t)k0s * NW;
            #pragma unroll
            for (int j = 0; j < 8; ++j) {
                const int p   = threadIdx.x + 256 * j;
                const int row = p >> 7;
                const int c4  = (p & 127) << 2;
                *(float4*)&Blds[buf][row][c4] = *(const float4*)(Bp + (size_t)row * NW + c4);
            }
        }
        __syncthreads();
#endif
        // ---- 4 WMMA k-steps over the chunk (8 x v_wmma_f32_16x16x4_f32 each) ----
        #pragma unroll
        for (int ks = 0; ks < KC; ks += 4) {
            const int kb = ks + 2 * half;
            v2f a0 = *(const v2f*)&Alds[buf][ln][kb];
            v2f a1 = *(const v2f*)&Alds[buf][16 + ln][kb];
            #pragma unroll
            for (int ns = 0; ns < 4; ++ns) {
                const int n = wave * 64 + ns * 16 + ln;
                v2f b;
                b.x = Blds[buf][kb][n];
                b.y = Blds[buf][kb + 1][n];
                acc[0][ns] = __builtin_amdgcn_wmma_f32_16x16x4_f32(
                    false, a0, false, b, (short)0, acc[0][ns], false, false);
                acc[1][ns] = __builtin_amdgcn_wmma_f32_16x16x4_f32(
                    false, a1, false, b, (short)0, acc[1][ns], false, false);
            }
        }
    }

    // ---- epilogue: (silu) + store ----
    #pragma unroll
    for (int ms = 0; ms < 2; ++ms)
        #pragma unroll
        for (int ns = 0; ns < 4; ++ns)
            #pragma unroll
            for (int r = 0; r < 8; ++r) {
                const int M   = ms * 16 + r + 8 * half;
                const int col = nblk + wave * 64 + ns * 16 + ln;
                float v = acc[ms][ns][r];
                if (SILU) v = v / (1.f + __expf(-v));
                Out[(size_t)(slotBase + M) * NW + col] = v;
            }
}

// =====================================================================
// Combine: out[t] = w0 * Y[slot0] + w1 * Y[slot1]   (float4 wide)
// =====================================================================
__global__ __launch_bounds__(256)
void moe_combine(const float* __restrict__ Y, const int* __restrict__ t2s,
                 const float* __restrict__ tw, float* __restrict__ out)
{
    const int idx = blockIdx.x * blockDim.x + threadIdx.x;   // over T_*Dm/4
    const int t  = idx / (Dm / 4);
    const int d4 = idx % (Dm / 4);
    const int   s0 = t2s[t * TOPK + 0], s1 = t2s[t * TOPK + 1];
    const float w0 = tw[t * TOPK + 0],  w1 = tw[t * TOPK + 1];
    const float4 y0 = ((const float4*)(Y + (size_t)s0 * Dm))[d4];
    const float4 y1 = ((const float4*)(Y + (size_t)s1 * Dm))[d4];
    float4 r;
    r.x = w0 * y0.x + w1 * y1.x;
    r.y = w0 * y0.y + w1 * y1.y;
    r.z = w0 * y0.z + w1 * y1.z;
    r.w = w0 * y0.w + w1 * y1.w;
    ((float4*)out)[idx] = r;
}

extern "C" void kernel_launch(void* const* d_in, const int* in_sizes, int n_in,
                              void* d_out, int out_size, void* d_ws, size_t ws_size,
                              hipStream_t stream)
{
    const float* hidden = (const float*)d_in[0];   // [T, D]
    const float* logits = (const float*)d_in[1];   // [T, E]
    const float* W1     = (const float*)d_in[2];   // [E, D, F]
    const float* W2     = (const float*)d_in[3];   // [E, F, D]
    float* out = (float*)d_out;                    // [T, D]

    char* ws = (char*)d_ws;
    int*   base = (int*)  (ws + WS_BASE);
    int*   s2t  = (int*)  (ws + WS_S2T);
    int*   t2s  = (int*)  (ws + WS_T2S);
    float* tw   = (float*)(ws + WS_TW);
    float* H    = (float*)(ws + WS_H);
    float* Y    = (float*)(ws + WS_Y);

    moe_router<<<dim3(1), dim3(1024), 0, stream>>>(logits, base, s2t, t2s, tw);

    dim3 g1(Ff / NTILE, MAX_MTILES);   // 4 x 136
    moe_gemm<Dm, Ff, true, true><<<g1, dim3(256), 0, stream>>>(hidden, W1, H, base, s2t);

    dim3 g2(Dm / NTILE, MAX_MTILES);   // 2 x 136
    moe_gemm<Ff, Dm, false, false><<<g2, dim3(256), 0, stream>>>(H, W2, Y, base, s2t);

    moe_combine<<<dim3(T_ * Dm / 4 / 256), dim3(256), 0, stream>>>(Y, t2s, tw, out);
}